// GraphNet_9938554323232
// MI455X (gfx1250) — compile-verified
//
#include <hip/hip_runtime.h>

#define N_NODES 100000
#define N_EDGES 1600000
#define GDIM 8
#define LDIM 128
#define FDIM 6
#define DT_STEP 0.01f
#define DIN (3 * LDIM + GDIM) /* 392 */

#define MTILE 32
#define LDI 404 /* padded stride for 392-wide input tile (20 mod 64 -> conflict-free b64) */
#define LDH 132 /* padded stride for 128-wide hidden tile */

typedef __attribute__((ext_vector_type(2))) float v2f;
typedef __attribute__((ext_vector_type(4))) float v4f;
typedef __attribute__((ext_vector_type(8))) float v8f;

// ---------------------------------------------------------------------------
// workspace layout (floats):
//   [0,N)        s_sum     [N,2N)   s_cnt    [2N,3N) r_sum   [3N,4N) r_cnt
//   [4N,132N)    h_n (N x 128)
//   [132N, ...)  packed W1 (3 x 392 x 128), then packed W2 (3 x 128 x 128)
// ---------------------------------------------------------------------------

__global__ void zero_ws_kernel(float* __restrict__ ws, int n) {
    int i = blockIdx.x * blockDim.x + threadIdx.x;
    if (i < n) ws[i] = 0.0f;
}

// Interleave weight row pairs so a lane's WMMA B-fragment {W[k][n], W[k+1][n]}
// is one contiguous b64:  Wp[(k/2)*2L + 2n + (k&1)] = W[k*L + n]
__global__ void pack_weights_kernel(const float* __restrict__ W1,
                                    const float* __restrict__ W2,
                                    float* __restrict__ W1p, float* __restrict__ W2p) {
    int i = blockIdx.x * blockDim.x + threadIdx.x;
    const int n1 = 3 * DIN * LDIM;
    const int n2 = 3 * LDIM * LDIM;
    if (i < n1) {
        int s = i / (DIN * LDIM);
        int rem = i - s * (DIN * LDIM);
        int k = rem / LDIM;
        int n = rem - k * LDIM;
        W1p[s * (DIN * LDIM) + (k >> 1) * (2 * LDIM) + 2 * n + (k & 1)] = W1[i];
    } else if (i < n1 + n2) {
        int j = i - n1;
        int s = j / (LDIM * LDIM);
        int rem = j - s * (LDIM * LDIM);
        int k = rem / LDIM;
        int n = rem - k * LDIM;
        W2p[s * (LDIM * LDIM) + (k >> 1) * (2 * LDIM) + 2 * n + (k & 1)] = W2[j];
    }
}

// h_n = nodes @ enc_node_w + enc_node_b   (N,6)@(6,128)
__global__ void encode_nodes_kernel(const float* __restrict__ nodes,
                                    const float* __restrict__ w,
                                    const float* __restrict__ b,
                                    float* __restrict__ hn) {
    int idx = blockIdx.x * blockDim.x + threadIdx.x;
    if (idx >= N_NODES * LDIM) return;
    int n = idx >> 7;
    int c = idx & (LDIM - 1);
    const float* nd = nodes + n * FDIM;
    float acc = b[c];
#pragma unroll
    for (int f = 0; f < FDIM; ++f) acc = fmaf(nd[f], w[f * LDIM + c], acc);
    hn[idx] = acc;
}

// rank-1 segment sums: S_n = sum(edges[e]), C_n = count, per sender/receiver
__global__ void edge_agg_kernel(const float* __restrict__ edges,
                                const int* __restrict__ senders,
                                const int* __restrict__ receivers,
                                float* __restrict__ s_sum, float* __restrict__ s_cnt,
                                float* __restrict__ r_sum, float* __restrict__ r_cnt) {
    int e = blockIdx.x * blockDim.x + threadIdx.x;
    if (e >= N_EDGES) return;
    float v = edges[e];
    int s = senders[e];
    int r = receivers[e];
    atomicAdd(&s_sum[s], v);
    atomicAdd(&s_cnt[s], 1.0f);
    atomicAdd(&r_sum[r], v);
    atomicAdd(&r_cnt[r], 1.0f);
}

// One message-passing step, per 32-node tile:
//   inp = [h_n | S*ew + C*eb (send) | S*ew + C*eb (recv) | g]  (32 x 392 in LDS)
//   h   = relu(inp @ W1 + b1);  h_n = h @ W2 + b2   — via V_WMMA_F32_16X16X4_F32
__global__ __launch_bounds__(256) void mlp_step_kernel(
    float* __restrict__ hn,
    const float* __restrict__ s_sum, const float* __restrict__ s_cnt,
    const float* __restrict__ r_sum, const float* __restrict__ r_cnt,
    const float* __restrict__ ew, const float* __restrict__ eb,
    const float* __restrict__ g,
    const float* __restrict__ W1p, const float* __restrict__ B1,
    const float* __restrict__ W2p, const float* __restrict__ B2,
    float g1_offset) {
    __shared__ __align__(16) float smem[MTILE * LDI]; // 51,712 B; reused for h after barrier

    const int m0 = blockIdx.x * MTILE;
    const int tid = threadIdx.x;

    // ---- phase 0: build 32x392 input tile in LDS (vectorized b128) ----
    {
        const v4f* hn4 = (const v4f*)(hn + (size_t)m0 * LDIM);
        for (int idx = tid; idx < MTILE * (LDIM / 4); idx += 256) {
            int row = idx >> 5; // 32 float4 per row
            int c4 = idx & 31;
            *(v4f*)&smem[row * LDI + 4 * c4] = hn4[row * 32 + c4];
        }
    }
    for (int idx = tid; idx < MTILE * 64; idx += 256) { // sent+recv: 64 float4 per row
        int row = idx >> 6;
        int c4 = idx & 63;
        int cc = c4 & 31;
        int n = m0 + row;
        v4f w = *(const v4f*)&ew[4 * cc];
        v4f bb = *(const v4f*)&eb[4 * cc];
        float sv, cv;
        if (c4 < 32) { sv = s_sum[n]; cv = s_cnt[n]; }
        else         { sv = r_sum[n]; cv = r_cnt[n]; }
        v4f v = w * sv + bb * cv;
        *(v4f*)&smem[row * LDI + LDIM + 4 * c4] = v;
    }
    if (tid < MTILE * GDIM) {
        int row = tid >> 3;
        int c = tid & 7;
        smem[row * LDI + 3 * LDIM + c] = g[c] + (c == 1 ? g1_offset : 0.0f);
    }
    __syncthreads();

    const int lane = tid & 31;
    const int wave = tid >> 5;  // 8 waves -> 8 column strips of 16
    const int half = lane >> 4; // K sub-phase for f32 WMMA fragments
    const int r = lane & 15;
    const int n0 = wave * 16;

    // invariant per-lane base pointers: packed index = k*LDIM + (half*2L + 2*(n0+r))
    const float* wb1 = W1p + half * (2 * LDIM) + 2 * (n0 + r);
    const float* wb2 = W2p + half * (2 * LDIM) + 2 * (n0 + r);
    const float* a0p = &smem[(0 + r) * LDI + 2 * half];
    const float* a1p = &smem[(16 + r) * LDI + 2 * half];

    // ---- layer 1: (32x392) @ (392x128), K-step 4, packed-b64 weight frags ----
    v8f c0 = {}; // rows m0..m0+15, cols n0..n0+15
    v8f c1 = {}; // rows m0+16..m0+31
    for (int k = 0; k < DIN; k += 4) {
        v2f a0 = *(const v2f*)(a0p + k);
        v2f a1 = *(const v2f*)(a1p + k);
        v2f b = *(const v2f*)(wb1 + k * LDIM); // byte offset k*512: folds into imm
        c0 = __builtin_amdgcn_wmma_f32_16x16x4_f32(false, a0, false, b, (short)0, c0, false, false);
        c1 = __builtin_amdgcn_wmma_f32_16x16x4_f32(false, a1, false, b, (short)0, c1, false, false);
    }
    __syncthreads(); // everyone done reading the input tile -> LDS reuse is safe

    const float bias1 = B1[n0 + r];
#pragma unroll
    for (int i = 0; i < 8; ++i) {
        int row = i + half * 8;
        smem[(row)*LDH + n0 + r] = fmaxf(c0[i] + bias1, 0.0f);
        smem[(row + 16) * LDH + n0 + r] = fmaxf(c1[i] + bias1, 0.0f);
    }
    __syncthreads();

    // ---- layer 2: (32x128) @ (128x128) ----
    const float* h0p = &smem[(0 + r) * LDH + 2 * half];
    const float* h1p = &smem[(16 + r) * LDH + 2 * half];
    v8f d0 = {};
    v8f d1 = {};
    for (int k = 0; k < LDIM; k += 4) {
        v2f a0 = *(const v2f*)(h0p + k);
        v2f a1 = *(const v2f*)(h1p + k);
        v2f b = *(const v2f*)(wb2 + k * LDIM);
        d0 = __builtin_amdgcn_wmma_f32_16x16x4_f32(false, a0, false, b, (short)0, d0, false, false);
        d1 = __builtin_amdgcn_wmma_f32_16x16x4_f32(false, a1, false, b, (short)0, d1, false, false);
    }
    const float bias2 = B2[n0 + r];
#pragma unroll
    for (int i = 0; i < 8; ++i) {
        int row = i + half * 8;
        hn[(m0 + row) * LDIM + n0 + r] = d0[i] + bias2;      // in-place: block owns rows
        hn[(m0 + row + 16) * LDIM + n0 + r] = d1[i] + bias2;
    }
}

// one wave per node: pred_acc = h_n . dec_w + dec_b; integrate; write 7-col node row
__global__ __launch_bounds__(256) void decode_kernel(const float* __restrict__ hn,
                                                     const float* __restrict__ nodes,
                                                     const float* __restrict__ dw,
                                                     const float* __restrict__ db,
                                                     float* __restrict__ out) {
    int node = (blockIdx.x * blockDim.x + threadIdx.x) >> 5;
    int lane = threadIdx.x & 31;
    if (node >= N_NODES) return;
    const float* h = hn + node * LDIM;
    float s = 0.0f;
#pragma unroll
    for (int j = 0; j < 4; ++j) s = fmaf(h[lane + 32 * j], dw[lane + 32 * j], s);
#pragma unroll
    for (int off = 16; off > 0; off >>= 1) s += __shfl_xor(s, off, 32);
    if (lane == 0) {
        float acc = s + db[0];
        const float* nd = nodes + node * FDIM;
        float next_vel = fmaf(acc, DT_STEP, nd[5]);
        float next_pos = fmaf(next_vel, DT_STEP, nd[0]);
        float* o = out + node * 7;
        o[0] = next_pos;
        o[1] = nd[1];
        o[2] = nd[2];
        o[3] = nd[3];
        o[4] = nd[4];
        o[5] = next_vel;
        o[6] = acc;
    }
}

// next_edge = diff(next_pos); append final globals
__global__ void finalize_kernel(float* __restrict__ out, const float* __restrict__ g) {
    int i = blockIdx.x * blockDim.x + threadIdx.x;
    if (i < N_NODES - 1) out[N_NODES * 7 + i] = out[(i + 1) * 7] - out[i * 7];
    if (i < GDIM) out[N_NODES * 7 + (N_NODES - 1) + i] = g[i] + (i == 1 ? 3.0f : 0.0f);
}

extern "C" void kernel_launch(void* const* d_in, const int* in_sizes, int n_in,
                              void* d_out, int out_size, void* d_ws, size_t ws_size,
                              hipStream_t stream) {
    (void)in_sizes; (void)n_in; (void)out_size; (void)ws_size;
    const float* nodes      = (const float*)d_in[0];
    const float* edges      = (const float*)d_in[1];
    const float* globals_   = (const float*)d_in[2];
    const int*   senders    = (const int*)d_in[3];
    const int*   receivers  = (const int*)d_in[4];
    const float* enc_node_w = (const float*)d_in[5];
    const float* enc_node_b = (const float*)d_in[6];
    const float* enc_edge_w = (const float*)d_in[7];
    const float* enc_edge_b = (const float*)d_in[8];
    const float* mlp_w1     = (const float*)d_in[9];
    const float* mlp_b1     = (const float*)d_in[10];
    const float* mlp_w2     = (const float*)d_in[11];
    const float* mlp_b2     = (const float*)d_in[12];
    const float* dec_w      = (const float*)d_in[13];
    const float* dec_b      = (const float*)d_in[14];

    float* out = (float*)d_out;
    float* ws = (float*)d_ws;
    float* s_sum = ws;
    float* s_cnt = ws + N_NODES;
    float* r_sum = ws + 2 * N_NODES;
    float* r_cnt = ws + 3 * N_NODES;
    float* hn    = ws + 4 * (size_t)N_NODES;            // N * 128 floats
    float* w1p   = ws + 132 * (size_t)N_NODES;          // 3 * 392 * 128
    float* w2p   = w1p + 3 * (size_t)DIN * LDIM;        // 3 * 128 * 128

    zero_ws_kernel<<<(4 * N_NODES + 255) / 256, 256, 0, stream>>>(ws, 4 * N_NODES);

    pack_weights_kernel<<<(3 * (DIN + LDIM) * LDIM + 255) / 256, 256, 0, stream>>>(
        mlp_w1, mlp_w2, w1p, w2p);

    encode_nodes_kernel<<<(N_NODES * LDIM + 255) / 256, 256, 0, stream>>>(
        nodes, enc_node_w, enc_node_b, hn);

    edge_agg_kernel<<<(N_EDGES + 255) / 256, 256, 0, stream>>>(
        edges, senders, receivers, s_sum, s_cnt, r_sum, r_cnt);

    for (int s = 0; s < 3; ++s) {
        mlp_step_kernel<<<N_NODES / MTILE, 256, 0, stream>>>(
            hn, s_sum, s_cnt, r_sum, r_cnt, enc_edge_w, enc_edge_b, globals_,
            w1p + (size_t)s * DIN * LDIM, mlp_b1 + (size_t)s * LDIM,
            w2p + (size_t)s * LDIM * LDIM, mlp_b2 + (size_t)s * LDIM, (float)s);
    }

    decode_kernel<<<(N_NODES * 32) / 256, 256, 0, stream>>>(hn, nodes, dec_w, dec_b, out);

    finalize_kernel<<<(N_NODES + 255) / 256, 256, 0, stream>>>(out, globals_);
}